// VpSdeGaussianAnalytical_45792941310157
// MI455X (gfx1250) — compile-verified
//
#include <hip/hip_runtime.h>
#include <math.h>

// ---------------------------------------------------------------------------
// VP-SDE Gaussian analytical score: batched solve of (C + s_b I) y = rhs_b
// via Chebyshev iteration (eigenvalue bounds known analytically), with the
// shared-matrix matvec expressed as a 512x512x512 fp32 GEMM on
// v_wmma_f32_16x16x4_f32, and (when available) async global->LDS double
// buffering on the CDNA5 ASYNCcnt path.
// ---------------------------------------------------------------------------

#define BATCH 512
#define DIM   512
#define NITER 21        // 21 total matvec applications (incl. d0) -> ~5e-9 residual

#define BM 64
#define BN 64
#define BK 32
#define LDSP (BK + 4)   // padded LDS row stride (floats), keeps b64 loads conflict-free

typedef float v8f __attribute__((ext_vector_type(8)));
typedef float v2f __attribute__((ext_vector_type(2)));

#if defined(__has_builtin)
#if __has_builtin(__builtin_amdgcn_global_load_async_to_lds_b128) && \
    __has_builtin(__builtin_amdgcn_s_wait_asynccnt)
#define USE_ASYNC_LDS 1
#endif
#endif
#ifndef USE_ASYNC_LDS
#define USE_ASYNC_LDS 0
#endif

#if USE_ASYNC_LDS
// Matches the builtin's parameter pointee type exactly:
//   '__attribute__((__vector_size__(4 * sizeof(int)))) int __device__ *'
typedef int v4i_vs __attribute__((vector_size(4 * sizeof(int))));
typedef __attribute__((address_space(1))) v4i_vs* g_b128_t;  // global (AS1)
typedef __attribute__((address_space(3))) v4i_vs* l_b128_t;  // LDS (AS3)

// Issue one 16-byte async global->LDS copy for this lane.
// Generic LDS pointer low 32 bits == LDS byte offset (flat-aperture rule).
__device__ __forceinline__ void async_copy16(const float* g, float* l) {
  g_b128_t gp = (g_b128_t)(uintptr_t)g;
  l_b128_t lp = (l_b128_t)(unsigned int)(uintptr_t)l;
  __builtin_amdgcn_global_load_async_to_lds_b128(gp, lp, 0, 0);
}
#endif

// ---------------------------------------------------------------------------
// Kernel 1: per-sample scalars + Chebyshev coefficient tables
//   B_t   = 9.95 t^2 + 0.1 t ; shift = e^B - 1 ; scale = e^{B/2}
//   oscale = -scale*sqrt(1-e^{-B})
//   spectrum of (C + shift I) in [1+shift, 6+shift] -> theta = 3.5+shift, delta = 2.5
// ---------------------------------------------------------------------------
__global__ void cheb_setup_scalars(const float* __restrict__ t,
                                   float* __restrict__ shift,
                                   float* __restrict__ oscale,
                                   float* __restrict__ invtheta,
                                   float* __restrict__ scaleArr,
                                   float* __restrict__ c1,
                                   float* __restrict__ c2) {
  int b = blockIdx.x * blockDim.x + threadIdx.x;
  if (b >= BATCH) return;
  float tb = t[b];
  float B  = 9.95f * tb * tb + 0.1f * tb;
  float s  = expm1f(B);
  float sc = expf(0.5f * B);
  float rs = sqrtf(fmaxf(1.0f - expf(-B), 0.0f));
  shift[b]    = s;
  scaleArr[b] = sc;
  oscale[b]   = -sc * rs;
  float theta = 3.5f + s;
  float delta = 2.5f;
  invtheta[b] = 1.0f / theta;
  float sigma = theta / delta;
  float rho_prev = 1.0f / sigma;
  for (int k = 1; k <= NITER; ++k) {
    float rho = 1.0f / (2.0f * sigma - rho_prev);
    c1[k * BATCH + b] = rho * rho_prev;      // d_k = c1*d_{k-1} + c2*r_k
    c2[k * BATCH + b] = 2.0f * rho / delta;
    rho_prev = rho;
  }
}

// ---------------------------------------------------------------------------
// Kernel 2: vector init.  r0 = scale_b*x - mu ;  d0 = r0/theta ;  x1 = d0
// ---------------------------------------------------------------------------
__global__ void cheb_setup_vectors(const float* __restrict__ xin,
                                   const float* __restrict__ mu,
                                   const float* __restrict__ scaleArr,
                                   const float* __restrict__ invtheta,
                                   float* __restrict__ rv,
                                   float* __restrict__ dv,
                                   float* __restrict__ xv) {
  int idx = blockIdx.x * blockDim.x + threadIdx.x;
  int b   = idx >> 9;       // / DIM
  int col = idx & (DIM - 1);
  float r0 = scaleArr[b] * xin[idx] - mu[col];
  float d0 = invtheta[b] * r0;
  rv[idx] = r0;
  dv[idx] = d0;
  xv[idx] = d0;
}

// ---------------------------------------------------------------------------
// Kernel 3: one Chebyshev iteration, fully fused.
//   t = (C + diag(shift)) d_old     -- WMMA fp32 GEMM (C symmetric: row loads only)
//   r <- r - t ; d_new = c1*d_old + c2*r ; x <- x + d_new
//   last iteration: out = oscale * x
// ---------------------------------------------------------------------------
__global__ __launch_bounds__(256)
void cheb_iter(const float* __restrict__ dIn,
               const float* __restrict__ Cmat,   // [DIM][DIM], symmetric
               float* __restrict__ dOut,
               float* __restrict__ rVec,
               float* __restrict__ xVec,
               const float* __restrict__ shift,
               const float* __restrict__ c1k,
               const float* __restrict__ c2k,
               const float* __restrict__ oscale,
               float* __restrict__ out,
               int writeOut) {
#if USE_ASYNC_LDS
  __shared__ float As[2][BM * LDSP];
  __shared__ float Bs[2][BN * LDSP];
#else
  __shared__ float As[1][BM * LDSP];
  __shared__ float Bs[1][BN * LDSP];
#endif

  const int tid  = threadIdx.x;
  const int lane = tid & 31;
  const int wave = tid >> 5;      // 0..7
  const int wm   = wave >> 1;     // 0..3 : wave M quadrant (16 rows)
  const int wn   = wave & 1;      // 0..1 : wave N half (32 cols)
  const int half = lane >> 4;     // K-pair selector within fragment
  const int l16  = lane & 15;
  const int mBase = blockIdx.y * BM;
  const int nBase = blockIdx.x * BN;

  v8f acc0 = {};
  v8f acc1 = {};

  // cooperative staging: 64 rows x 32 floats per matrix per K step
  const int lrow = tid >> 2;      // 0..63
  const int lc4  = tid & 3;       // float4 column group

  const float* aRow = dIn  + (size_t)(mBase + lrow) * DIM;
  const float* bRow = Cmat + (size_t)(nBase + lrow) * DIM;  // symmetry: C[i][k] rows

#if USE_ASYNC_LDS
  // -------- async double-buffered pipeline (4 in-flight b128 per wave/group) --
  {
    // prologue: buffer 0 <- K chunk 0
    async_copy16(aRow + 0 + lc4 * 4,      &As[0][lrow * LDSP + lc4 * 4]);
    async_copy16(aRow + 0 + lc4 * 4 + 16, &As[0][lrow * LDSP + lc4 * 4 + 16]);
    async_copy16(bRow + 0 + lc4 * 4,      &Bs[0][lrow * LDSP + lc4 * 4]);
    async_copy16(bRow + 0 + lc4 * 4 + 16, &Bs[0][lrow * LDSP + lc4 * 4 + 16]);
  }
  int cur = 0;
  for (int k0 = 0; k0 < DIM; k0 += BK) {
    if (k0 + BK < DIM) {
      int nxt = cur ^ 1;
      async_copy16(aRow + (k0 + BK) + lc4 * 4,      &As[nxt][lrow * LDSP + lc4 * 4]);
      async_copy16(aRow + (k0 + BK) + lc4 * 4 + 16, &As[nxt][lrow * LDSP + lc4 * 4 + 16]);
      async_copy16(bRow + (k0 + BK) + lc4 * 4,      &Bs[nxt][lrow * LDSP + lc4 * 4]);
      async_copy16(bRow + (k0 + BK) + lc4 * 4 + 16, &Bs[nxt][lrow * LDSP + lc4 * 4 + 16]);
      __builtin_amdgcn_s_wait_asynccnt(4);  // previous group (4 ops) done; next in flight
    } else {
      __builtin_amdgcn_s_wait_asynccnt(0);  // last tile: everything done
    }
    __syncthreads();                         // all waves' tiles visible in LDS

    const float* AsC = &As[cur][0];
    const float* BsC = &Bs[cur][0];
#pragma unroll
    for (int kk = 0; kk < BK; kk += 4) {
      v2f af  = *reinterpret_cast<const v2f*>(&AsC[(wm * 16 + l16) * LDSP + kk + 2 * half]);
      v2f bf0 = *reinterpret_cast<const v2f*>(&BsC[(wn * 32 + l16) * LDSP + kk + 2 * half]);
      v2f bf1 = *reinterpret_cast<const v2f*>(&BsC[(wn * 32 + 16 + l16) * LDSP + kk + 2 * half]);
      acc0 = __builtin_amdgcn_wmma_f32_16x16x4_f32(false, af, false, bf0,
                                                   (short)0, acc0, false, false);
      acc1 = __builtin_amdgcn_wmma_f32_16x16x4_f32(false, af, false, bf1,
                                                   (short)0, acc1, false, false);
    }
    __syncthreads();   // all waves done reading 'cur' before it is overwritten
    cur ^= 1;
  }
#else
  // -------- fallback: register-staged single-buffer pipeline -----------------
  for (int k0 = 0; k0 < DIM; k0 += BK) {
    float4 a0 = reinterpret_cast<const float4*>(aRow + k0)[lc4];
    float4 a1 = reinterpret_cast<const float4*>(aRow + k0)[lc4 + 4];
    float4 b0 = reinterpret_cast<const float4*>(bRow + k0)[lc4];
    float4 b1 = reinterpret_cast<const float4*>(bRow + k0)[lc4 + 4];
    __syncthreads();  // WAR: previous inner loop finished reading LDS
    *reinterpret_cast<float4*>(&As[0][lrow * LDSP + lc4 * 4])      = a0;
    *reinterpret_cast<float4*>(&As[0][lrow * LDSP + lc4 * 4 + 16]) = a1;
    *reinterpret_cast<float4*>(&Bs[0][lrow * LDSP + lc4 * 4])      = b0;
    *reinterpret_cast<float4*>(&Bs[0][lrow * LDSP + lc4 * 4 + 16]) = b1;
    __syncthreads();

#pragma unroll
    for (int kk = 0; kk < BK; kk += 4) {
      v2f af  = *reinterpret_cast<const v2f*>(&As[0][(wm * 16 + l16) * LDSP + kk + 2 * half]);
      v2f bf0 = *reinterpret_cast<const v2f*>(&Bs[0][(wn * 32 + l16) * LDSP + kk + 2 * half]);
      v2f bf1 = *reinterpret_cast<const v2f*>(&Bs[0][(wn * 32 + 16 + l16) * LDSP + kk + 2 * half]);
      acc0 = __builtin_amdgcn_wmma_f32_16x16x4_f32(false, af, false, bf0,
                                                   (short)0, acc0, false, false);
      acc1 = __builtin_amdgcn_wmma_f32_16x16x4_f32(false, af, false, bf1,
                                                   (short)0, acc1, false, false);
    }
  }
#endif

  // Fused Chebyshev epilogue. C/D layout: VGPR v, lanes 0-15 -> M=v, lanes 16-31 -> M=v+8.
  const int rowB = mBase + wm * 16 + half * 8;
  const int col0 = nBase + wn * 32 + l16;
  const int col1 = col0 + 16;
#pragma unroll
  for (int v = 0; v < 8; ++v) {
    int row  = rowB + v;
    float sh = shift[row];
    float a1c = c1k[row];
    float a2c = c2k[row];
    size_t i0 = (size_t)row * DIM + col0;
    size_t i1 = (size_t)row * DIM + col1;
    float dold0 = dIn[i0], dold1 = dIn[i1];
    float t0 = acc0[v] + sh * dold0;        // + diag(shift) part of the operator
    float t1 = acc1[v] + sh * dold1;
    float rn0 = rVec[i0] - t0;
    float rn1 = rVec[i1] - t1;
    float dn0 = a1c * dold0 + a2c * rn0;
    float dn1 = a1c * dold1 + a2c * rn1;
    float xn0 = xVec[i0] + dn0;
    float xn1 = xVec[i1] + dn1;
    rVec[i0] = rn0;  rVec[i1] = rn1;
    dOut[i0] = dn0;  dOut[i1] = dn1;
    xVec[i0] = xn0;  xVec[i1] = xn1;
    if (writeOut) {
      float os = oscale[row];
      out[i0] = os * xn0;
      out[i1] = os * xn1;
    }
  }
}

// ---------------------------------------------------------------------------
extern "C" void kernel_launch(void* const* d_in, const int* in_sizes, int n_in,
                              void* d_out, int out_size, void* d_ws, size_t ws_size,
                              hipStream_t stream) {
  const float* x  = (const float*)d_in[0];   // [512,512]
  const float* t  = (const float*)d_in[1];   // [512]
  const float* mu = (const float*)d_in[2];   // [512]
  const float* C  = (const float*)d_in[3];   // [512,512] SPD
  float* out = (float*)d_out;
  float* ws  = (float*)d_ws;

  const size_t NB = (size_t)BATCH * DIM;
  float* d0 = ws;                // ping
  float* d1 = ws + NB;           // pong
  float* rv = ws + 2 * NB;
  float* xv = ws + 3 * NB;
  float* shift    = ws + 4 * NB;
  float* oscale   = shift + BATCH;
  float* invtheta = oscale + BATCH;
  float* scaleArr = invtheta + BATCH;
  float* c1 = scaleArr + BATCH;
  float* c2 = c1 + (NITER + 1) * BATCH;
  (void)in_sizes; (void)n_in; (void)out_size; (void)ws_size;

  cheb_setup_scalars<<<2, 256, 0, stream>>>(t, shift, oscale, invtheta, scaleArr, c1, c2);
  cheb_setup_vectors<<<(BATCH * DIM) / 256, 256, 0, stream>>>(x, mu, scaleArr, invtheta,
                                                              rv, d0, xv);
  dim3 grid(DIM / BN, BATCH / BM);  // 8 x 8 = 64 workgroups
  for (int k = 1; k <= NITER; ++k) {
    const float* din = (k & 1) ? d0 : d1;
    float*      dout = (k & 1) ? d1 : d0;
    cheb_iter<<<grid, 256, 0, stream>>>(din, C, dout, rv, xv, shift,
                                        c1 + (size_t)k * BATCH,
                                        c2 + (size_t)k * BATCH,
                                        oscale, out, (k == NITER) ? 1 : 0);
  }
}